// GeometricUpdateHead_40200893891378
// MI455X (gfx1250) — compile-verified
//
#include <hip/hip_runtime.h>
#include <hip/hip_bf16.h>
#include <math.h>

#define D_MODEL 1024
#define NPTS    8192
#define NBATCH  8
#define NMOD    8
#define MROWS   (NBATCH * NPTS)      // 65536
#define PCOLS   (NMOD * 8)           // 64
#define PI_F    3.14159265358979323846f
#define MAX_T   3.0f

typedef __attribute__((ext_vector_type(2))) float v2f;
typedef __attribute__((ext_vector_type(8))) float v8f;

// ---------------------------------------------------------------------------
// Kernel 1: params = node_feats @ W.T + b   (M=65536, N=64, K=1024, fp32 WMMA)
// Each wave: 32 rows x 64 cols -> 8 accumulators; every W fragment reused by
// two WMMAs (6 loads : 8 wmma). Block = 8 waves = 256 rows.
// ---------------------------------------------------------------------------
__global__ void __launch_bounds__(256)
gemm_params_kernel(const float* __restrict__ A,     // (65536, 1024)
                   const float* __restrict__ W,     // (64, 1024)
                   const float* __restrict__ bias,  // (64)
                   float* __restrict__ P)           // (65536, 64)
{
    const int lane    = threadIdx.x & 31;
    const int wave    = threadIdx.x >> 5;
    const int mBase   = blockIdx.x * 256 + wave * 32;
    const int r       = lane & 15;
    const int halfSel = lane >> 4;       // 0: K pair {k0,k0+1}, 1: {k0+2,k0+3}
    const int koff    = halfSel * 2;

    const float* aRow0 = A + (size_t)(mBase + r) * D_MODEL + koff;
    const float* aRow1 = A + (size_t)(mBase + 16 + r) * D_MODEL + koff;
    const float* w0    = W + (size_t)(0 * 16 + r) * D_MODEL + koff;
    const float* w1    = W + (size_t)(1 * 16 + r) * D_MODEL + koff;
    const float* w2    = W + (size_t)(2 * 16 + r) * D_MODEL + koff;
    const float* w3    = W + (size_t)(3 * 16 + r) * D_MODEL + koff;

    __builtin_prefetch(w0, 0, 3);
    __builtin_prefetch(w1, 0, 3);
    __builtin_prefetch(w2, 0, 3);
    __builtin_prefetch(w3, 0, 3);

    v8f acc00 = {}, acc10 = {}, acc20 = {}, acc30 = {};   // rows 0-15
    v8f acc01 = {}, acc11 = {}, acc21 = {}, acc31 = {};   // rows 16-31

    for (int k0 = 0; k0 < D_MODEL; k0 += 4) {
        v2f a0 = *(const v2f*)(aRow0 + k0);
        v2f a1 = *(const v2f*)(aRow1 + k0);
        v2f b0 = *(const v2f*)(w0 + k0);
        v2f b1 = *(const v2f*)(w1 + k0);
        v2f b2 = *(const v2f*)(w2 + k0);
        v2f b3 = *(const v2f*)(w3 + k0);
        acc00 = __builtin_amdgcn_wmma_f32_16x16x4_f32(false, a0, false, b0, (short)0, acc00, false, false);
        acc01 = __builtin_amdgcn_wmma_f32_16x16x4_f32(false, a1, false, b0, (short)0, acc01, false, false);
        acc10 = __builtin_amdgcn_wmma_f32_16x16x4_f32(false, a0, false, b1, (short)0, acc10, false, false);
        acc11 = __builtin_amdgcn_wmma_f32_16x16x4_f32(false, a1, false, b1, (short)0, acc11, false, false);
        acc20 = __builtin_amdgcn_wmma_f32_16x16x4_f32(false, a0, false, b2, (short)0, acc20, false, false);
        acc21 = __builtin_amdgcn_wmma_f32_16x16x4_f32(false, a1, false, b2, (short)0, acc21, false, false);
        acc30 = __builtin_amdgcn_wmma_f32_16x16x4_f32(false, a0, false, b3, (short)0, acc30, false, false);
        acc31 = __builtin_amdgcn_wmma_f32_16x16x4_f32(false, a1, false, b3, (short)0, acc31, false, false);
    }

    // D layout: VGPR rr -> M = halfSel*8 + rr, N = jtile*16 + (lane & 15)
    const float bv0 = bias[0 * 16 + r];
    const float bv1 = bias[1 * 16 + r];
    const float bv2 = bias[2 * 16 + r];
    const float bv3 = bias[3 * 16 + r];
#pragma unroll
    for (int rr = 0; rr < 8; ++rr) {
        const size_t row0 = (size_t)(mBase + halfSel * 8 + rr) * PCOLS;
        const size_t row1 = (size_t)(mBase + 16 + halfSel * 8 + rr) * PCOLS;
        P[row0 + 0 * 16 + r] = acc00[rr] + bv0;
        P[row0 + 1 * 16 + r] = acc10[rr] + bv1;
        P[row0 + 2 * 16 + r] = acc20[rr] + bv2;
        P[row0 + 3 * 16 + r] = acc30[rr] + bv3;
        P[row1 + 0 * 16 + r] = acc01[rr] + bv0;
        P[row1 + 1 * 16 + r] = acc11[rr] + bv1;
        P[row1 + 2 * 16 + r] = acc21[rr] + bv2;
        P[row1 + 3 * 16 + r] = acc31[rr] + bv3;
    }
}

// ---------------------------------------------------------------------------
// Wave32 reductions
// ---------------------------------------------------------------------------
__device__ __forceinline__ float waveSum(float v) {
#pragma unroll
    for (int o = 16; o > 0; o >>= 1) v += __shfl_xor(v, o, 32);
    return v;
}
__device__ __forceinline__ float waveMax(float v) {
#pragma unroll
    for (int o = 16; o > 0; o >>= 1) v = fmaxf(v, __shfl_xor(v, o, 32));
    return v;
}

// ---------------------------------------------------------------------------
// Kernel 2: per-(b,m) softmax stats of anchor logits over N=8192
// grid = 64 blocks (b*8+m), block = 1024 (32 wave32s)
// ---------------------------------------------------------------------------
__global__ void __launch_bounds__(1024)
softmax_stats_kernel(const float* __restrict__ P, float* __restrict__ stats)
{
    const int b = blockIdx.x >> 3;
    const int m = blockIdx.x & 7;
    __shared__ float red[32];
    __shared__ float bcast;

    const int wave = threadIdx.x >> 5;
    const int lane = threadIdx.x & 31;

    // pass 1: max
    float mx = -3.402823466e38f;
#pragma unroll
    for (int i = 0; i < NPTS / 1024; ++i) {
        const int n = threadIdx.x + i * 1024;
        mx = fmaxf(mx, P[((size_t)(b * NPTS + n)) * PCOLS + m * 8]);
    }
    mx = waveMax(mx);
    if (lane == 0) red[wave] = mx;
    __syncthreads();
    if (threadIdx.x == 0) {
        float t = red[0];
#pragma unroll
        for (int w = 1; w < 32; ++w) t = fmaxf(t, red[w]);
        bcast = t;
    }
    __syncthreads();
    const float MX = bcast;
    __syncthreads();

    // pass 2: sum(exp(x - MX))
    float s = 0.0f;
#pragma unroll
    for (int i = 0; i < NPTS / 1024; ++i) {
        const int n = threadIdx.x + i * 1024;
        s += expf(P[((size_t)(b * NPTS + n)) * PCOLS + m * 8] - MX);
    }
    s = waveSum(s);
    if (lane == 0) red[wave] = s;
    __syncthreads();
    if (threadIdx.x == 0) {
        float t = 0.0f;
#pragma unroll
        for (int w = 0; w < 32; ++w) t += red[w];
        stats[blockIdx.x * 2 + 0] = MX;
        stats[blockIdx.x * 2 + 1] = t;
    }
}

// ---------------------------------------------------------------------------
// Per-point module parameters
// ---------------------------------------------------------------------------
struct PP { float aw, sp, qw, qx, qy, qz, tx, ty, tz; };

__device__ __forceinline__ PP computePoint(const float* __restrict__ p,
                                           float mx, float invSe)
{
    const float4 p0 = *(const float4*)(p);       // anchor, select, rx, ry
    const float4 p1 = *(const float4*)(p + 4);   // rz, t0, t1, t2
    PP o;
    o.aw = expf(p0.x - mx) * invSe;
    o.sp = 1.0f / (1.0f + expf(-p0.y));
    const float rx = p0.z, ry = p0.w, rz = p1.x;
    const float nrm  = sqrtf(rx * rx + ry * ry + rz * rz);
    const float inv  = 1.0f / fmaxf(nrm, 1e-12f);
    const float ang  = tanhf(nrm) * PI_F;
    const float half = 0.5f * ang;
    const float shoa = (fabsf(ang) < 1e-6f) ? (0.5f - ang * ang * (1.0f / 48.0f))
                                            : (sinf(half) / ang);
    const float sc = inv * ang * shoa;
    o.qw = cosf(half);
    o.qx = rx * sc; o.qy = ry * sc; o.qz = rz * sc;
    o.tx = tanhf(p1.y) * MAX_T;
    o.ty = tanhf(p1.z) * MAX_T;
    o.tz = tanhf(p1.w) * MAX_T;
    return o;
}

// ---------------------------------------------------------------------------
// Kernel 3 (x8, sequential): one module step of the scan.
// grid = 8 (one block per batch), block = 1024 (32 wave32s);
// block owns all N=8192 points of its batch -> race-free reduce/apply.
// ---------------------------------------------------------------------------
__global__ void __launch_bounds__(1024)
module_step_kernel(const float* __restrict__ P, const float* __restrict__ stats,
                   const float* __restrict__ src, float* __restrict__ dst, int m)
{
    const int b    = blockIdx.x;
    const int wave = threadIdx.x >> 5;
    const int lane = threadIdx.x & 31;

    __shared__ float red[32][11];
    __shared__ float tot[11];

    const float mx    = stats[(b * NMOD + m) * 2 + 0];
    const float invSe = 1.0f / stats[(b * NMOD + m) * 2 + 1];

    // ---- reduce phase: 11 accumulators over N ----
    float acc[11];
#pragma unroll
    for (int k = 0; k < 11; ++k) acc[k] = 0.0f;

#pragma unroll
    for (int i = 0; i < NPTS / 1024; ++i) {
        const int n = threadIdx.x + i * 1024;
        const size_t row = (size_t)(b * NPTS + n);
        const PP pp = computePoint(P + row * PCOLS + m * 8, mx, invSe);
        const float cx = src[row * 3 + 0];
        const float cy = src[row * 3 + 1];
        const float cz = src[row * 3 + 2];
        acc[0] += cx * pp.aw;  acc[1] += cy * pp.aw;  acc[2] += cz * pp.aw;
        acc[3] += pp.sp;
        acc[4] += pp.qw * pp.sp; acc[5] += pp.qx * pp.sp;
        acc[6] += pp.qy * pp.sp; acc[7] += pp.qz * pp.sp;
        acc[8] += pp.tx * pp.sp; acc[9] += pp.ty * pp.sp; acc[10] += pp.tz * pp.sp;
    }
#pragma unroll
    for (int k = 0; k < 11; ++k) {
        const float s = waveSum(acc[k]);
        if (lane == 0) red[wave][k] = s;
    }
    __syncthreads();
    if (threadIdx.x < 11) {
        float s = 0.0f;
#pragma unroll
        for (int w = 0; w < 32; ++w) s += red[w][threadIdx.x];
        tot[threadIdx.x] = s;
    }
    __syncthreads();

    // ---- group values (redundant per thread) ----
    const float ax = tot[0], ay = tot[1], az = tot[2];
    const float S  = fmaxf(tot[3], 1e-8f);
    float gw = tot[4] / S, gx = tot[5] / S, gy = tot[6] / S, gz = tot[7] / S;
    const float qn = 1.0f / fmaxf(sqrtf(gw * gw + gx * gx + gy * gy + gz * gz), 1e-12f);
    gw *= qn; gx *= qn; gy *= qn; gz *= qn;
    const float tgx = tot[8] / S, tgy = tot[9] / S, tgz = tot[10] / S;
    __syncthreads();

    // ---- apply phase ----
#pragma unroll
    for (int i = 0; i < NPTS / 1024; ++i) {
        const int n = threadIdx.x + i * 1024;
        const size_t row = (size_t)(b * NPTS + n);
        const PP pp = computePoint(P + row * PCOLS + m * 8, mx, invSe);
        const float cx = src[row * 3 + 0];
        const float cy = src[row * 3 + 1];
        const float cz = src[row * 3 + 2];
        const float dx = cx - ax, dy = cy - ay, dz = cz - az;
        // r1 = g * (0, d)
        const float w1 = -gx * dx - gy * dy - gz * dz;
        const float x1 =  gw * dx + gy * dz - gz * dy;
        const float y1 =  gw * dy - gx * dz + gz * dx;
        const float z1 =  gw * dz + gx * dy - gy * dx;
        // r2 = r1 * conj(g)  (vector part)
        const float ox = -w1 * gx + x1 * gw - y1 * gz + z1 * gy;
        const float oy = -w1 * gy + x1 * gz + y1 * gw - z1 * gx;
        const float oz = -w1 * gz - x1 * gy + y1 * gx + z1 * gw;
        const float mxx = ox + ax + tgx;
        const float myy = oy + ay + tgy;
        const float mzz = oz + az + tgz;
        dst[row * 3 + 0] = cx + pp.sp * (mxx - cx);
        dst[row * 3 + 1] = cy + pp.sp * (myy - cy);
        dst[row * 3 + 2] = cz + pp.sp * (mzz - cz);
    }
}

// ---------------------------------------------------------------------------
extern "C" void kernel_launch(void* const* d_in, const int* in_sizes, int n_in,
                              void* d_out, int out_size, void* d_ws, size_t ws_size,
                              hipStream_t stream) {
    (void)in_sizes; (void)n_in; (void)out_size; (void)ws_size;
    const float* node_feats = (const float*)d_in[0];   // (8, 8192, 1024)
    const float* coords     = (const float*)d_in[1];   // (8, 8192, 3)
    const float* W          = (const float*)d_in[2];   // (64, 1024)
    const float* bias       = (const float*)d_in[3];   // (64)
    float* out = (float*)d_out;                        // (8, 8192, 3)

    float* P     = (float*)d_ws;                       // 65536 x 64 fp32 = 16 MB
    float* stats = P + (size_t)MROWS * PCOLS;          // 64 x 2 fp32

    gemm_params_kernel<<<MROWS / 256, 256, 0, stream>>>(node_feats, W, bias, P);
    softmax_stats_kernel<<<NBATCH * NMOD, 1024, 0, stream>>>(P, stats);
    for (int m = 0; m < NMOD; ++m) {
        const float* src = (m == 0) ? coords : out;
        module_step_kernel<<<NBATCH, 1024, 0, stream>>>(P, stats, src, out, m);
    }
}